// SimpleGCN_34153579938070
// MI455X (gfx1250) — compile-verified
//
#include <hip/hip_runtime.h>

typedef __attribute__((ext_vector_type(2))) float v2f;
typedef __attribute__((ext_vector_type(8))) float v8f;

#define TPB 256
#define WAVES_PER_BLOCK 8

// ---------------------------------------------------------------- utilities
__global__ void k_init_deg(float* __restrict__ deg, int n) {
  int i = blockIdx.x * blockDim.x + threadIdx.x;
  if (i < n) deg[i] = 1.0f;  // self-loop contributes 1 to every degree
}

__global__ void k_edge_deg(const int* __restrict__ dst, float* __restrict__ deg, int e) {
  int i = blockIdx.x * blockDim.x + threadIdx.x;
  if (i < e) atomicAdd(&deg[dst[i]], 1.0f);
}

__global__ void k_dinv(float* __restrict__ deg, int n) {
  int i = blockIdx.x * blockDim.x + threadIdx.x;
  if (i < n) deg[i] = rsqrtf(deg[i]);  // deg >= 1 always (self-loops)
}

__global__ void k_norm(const int* __restrict__ src, const int* __restrict__ dst,
                       const float* __restrict__ dinv, float* __restrict__ norm, int e) {
  int i = blockIdx.x * blockDim.x + threadIdx.x;
  if (i < e) norm[i] = dinv[src[i]] * dinv[dst[i]];
}

__global__ void k_zero4(float* __restrict__ p, long long n4) {
  long long i = (long long)blockIdx.x * blockDim.x + threadIdx.x;
  if (i < n4) ((float4*)p)[i] = make_float4(0.f, 0.f, 0.f, 0.f);
}

// ---------------------------------------------------------------- WMMA GEMM
// H[M, Nout] = X[M, K] @ W[Nout, K]^T   (full fp32 via V_WMMA_F32_16X16X4_F32)
// One wave computes a 16x16 tile of H; 8 waves per block share one 16-row
// W tile staged in LDS (rows padded by 2 floats to avoid bank conflicts).
template <int K>
__global__ __launch_bounds__(TPB) void k_gemm_wmma(const float* __restrict__ X,
                                                   const float* __restrict__ W,
                                                   float* __restrict__ H,
                                                   int M, int Nout) {
  __shared__ float wtile[16 * (K + 2)];
  const int tid = threadIdx.x;
  const int nBase = blockIdx.x * 16;

  // cooperative stage of W tile (16 x K) into LDS, float2 granularity
  for (int idx = tid; idx < (16 * K) / 2; idx += TPB) {
    int r  = idx / (K / 2);
    int k2 = idx - r * (K / 2);
    *(float2*)&wtile[r * (K + 2) + k2 * 2] =
        *(const float2*)&W[(size_t)(nBase + r) * K + k2 * 2];
  }
  __syncthreads();

  const int wave  = tid >> 5;
  const int lane  = tid & 31;
  const int mTile = blockIdx.y * WAVES_PER_BLOCK + wave;
  if (mTile * 16 >= M) return;  // wave-uniform: EXEC stays all-ones for WMMA

  const int mBase = mTile * 16;
  const int rc    = lane & 15;        // A row index == B column index
  const int half  = lane >> 4;        // selects K-pair {0,1} vs {2,3}
  const int kOff  = half * 2;

  const float* xrow = X + (size_t)(mBase + rc) * K;
  const float* wrow = &wtile[rc * (K + 2)];

  // warm L2/L0 with the tail of this A row while the head streams
  __builtin_prefetch(xrow + K / 2, 0, 0);

  v8f acc = {0.f, 0.f, 0.f, 0.f, 0.f, 0.f, 0.f, 0.f};
#pragma unroll 8
  for (int kk = 0; kk < K; kk += 4) {
    float2 af = *(const float2*)&xrow[kk + kOff];
    float2 bf = *(const float2*)&wrow[kk + kOff];
    v2f a = {af.x, af.y};
    v2f b = {bf.x, bf.y};
    acc = __builtin_amdgcn_wmma_f32_16x16x4_f32(false, a, false, b,
                                                (short)0, acc, false, false);
  }

  // D layout: VGPR r, lanes 0-15 -> M=r, lanes 16-31 -> M=r+8; N = lane&15
#pragma unroll
  for (int r = 0; r < 8; ++r) {
    H[(size_t)(mBase + r + 8 * half) * Nout + nBase + rc] = acc[r];
  }
}

// ---------------------------------------------------------------- edge scatter
// out[dst] += h[src] * norm ; one thread per (edge, 4-channel chunk)
template <int C>
__global__ void k_edge_agg(const float* __restrict__ h, const float* __restrict__ norm,
                           const int* __restrict__ src, const int* __restrict__ dst,
                           float* __restrict__ out, int nEdges) {
  constexpr int CH4 = C / 4;
  long long t = (long long)blockIdx.x * blockDim.x + threadIdx.x;
  long long total = (long long)nEdges * CH4;
  if (t >= total) return;
  int e = (int)(t / CH4);
  int c = (int)(t - (long long)e * CH4) * 4;
  int s = src[e], d = dst[e];
  float w = norm[e];
  const float4 v = *(const float4*)&h[(size_t)s * C + c];
  float* o = &out[(size_t)d * C + c];
  atomicAdd(o + 0, v.x * w);
  atomicAdd(o + 1, v.y * w);
  atomicAdd(o + 2, v.z * w);
  atomicAdd(o + 3, v.w * w);
}

// ---------------------------------------------------------------- epilogues
// acc[i,c] += h[i,c]*dinv[i]^2 (self-loop) + bias[c] ; optional ReLU
template <int C, bool RELU>
__global__ void k_node_finish(float* __restrict__ acc, const float* __restrict__ h,
                              const float* __restrict__ dinv, const float* __restrict__ bias,
                              int nNodes) {
  constexpr int CH4 = C / 4;
  long long t = (long long)blockIdx.x * blockDim.x + threadIdx.x;
  if (t >= (long long)nNodes * CH4) return;
  int i = (int)(t / CH4);
  int c = (int)(t - (long long)i * CH4) * 4;
  float di = dinv[i];
  float sl = di * di;
  float4 a  = *(float4*)&acc[(size_t)i * C + c];
  float4 hv = *(const float4*)&h[(size_t)i * C + c];
  float4 bv = *(const float4*)&bias[c];
  a.x += hv.x * sl + bv.x;
  a.y += hv.y * sl + bv.y;
  a.z += hv.z * sl + bv.z;
  a.w += hv.w * sl + bv.w;
  if (RELU) {
    a.x = fmaxf(a.x, 0.f); a.y = fmaxf(a.y, 0.f);
    a.z = fmaxf(a.z, 0.f); a.w = fmaxf(a.w, 0.f);
  }
  *(float4*)&acc[(size_t)i * C + c] = a;
}

// ---------------------------------------------------------------- driver
extern "C" void kernel_launch(void* const* d_in, const int* in_sizes, int n_in,
                              void* d_out, int out_size, void* d_ws, size_t ws_size,
                              hipStream_t stream) {
  constexpr int IN_C = 128, HID_C = 256, OUT_C = 128;

  const float* x  = (const float*)d_in[0];
  const int*   ei = (const int*)d_in[1];
  const float* W1 = (const float*)d_in[2];
  const float* b1 = (const float*)d_in[3];
  const float* W2 = (const float*)d_in[4];
  const float* b2 = (const float*)d_in[5];
  float* out = (float*)d_out;

  const int N = in_sizes[0] / IN_C;  // 50000
  const int E = in_sizes[1] / 2;     // 800000
  const int* src = ei;
  const int* dst = ei + E;

  // workspace layout (floats)
  float* ws   = (float*)d_ws;
  float* dinv = ws;                                        // N
  float* norm = ws + (((size_t)N + 255) & ~(size_t)255);   // E
  float* h    = norm + (((size_t)E + 255) & ~(size_t)255); // N*HID_C (reused for layer2 N*OUT_C)
  float* agg  = h + (size_t)N * HID_C;                     // N*HID_C

  const int mTiles  = (N + 15) / 16;          // 3125
  const dim3 gemmBlk(TPB);
  const dim3 gemm1Grid(HID_C / 16, (mTiles + WAVES_PER_BLOCK - 1) / WAVES_PER_BLOCK);
  const dim3 gemm2Grid(OUT_C / 16, (mTiles + WAVES_PER_BLOCK - 1) / WAVES_PER_BLOCK);

  // ---- normalization: deg -> dinv -> per-edge norm
  k_init_deg<<<(N + TPB - 1) / TPB, TPB, 0, stream>>>(dinv, N);
  k_edge_deg<<<(E + TPB - 1) / TPB, TPB, 0, stream>>>(dst, dinv, E);
  k_dinv<<<(N + TPB - 1) / TPB, TPB, 0, stream>>>(dinv, N);
  k_norm<<<(E + TPB - 1) / TPB, TPB, 0, stream>>>(src, dst, dinv, norm, E);

  // ---- layer 1: h = x @ W1^T ; agg = scatter(h*norm) ; agg = relu(agg + self + b1)
  {
    long long n4 = (long long)N * HID_C / 4;
    k_zero4<<<(unsigned)((n4 + TPB - 1) / TPB), TPB, 0, stream>>>(agg, n4);
  }
  k_gemm_wmma<IN_C><<<gemm1Grid, gemmBlk, 0, stream>>>(x, W1, h, N, HID_C);
  {
    long long total = (long long)E * (HID_C / 4);
    k_edge_agg<HID_C><<<(unsigned)((total + TPB - 1) / TPB), TPB, 0, stream>>>(
        h, norm, src, dst, agg, E);
  }
  {
    long long total = (long long)N * (HID_C / 4);
    k_node_finish<HID_C, true><<<(unsigned)((total + TPB - 1) / TPB), TPB, 0, stream>>>(
        agg, h, dinv, b1, N);
  }

  // ---- layer 2: h = agg @ W2^T ; out = scatter(h*norm) + self + b2
  {
    long long n4 = (long long)N * OUT_C / 4;
    k_zero4<<<(unsigned)((n4 + TPB - 1) / TPB), TPB, 0, stream>>>(out, n4);
  }
  k_gemm_wmma<HID_C><<<gemm2Grid, gemmBlk, 0, stream>>>(agg, W2, h, N, OUT_C);
  {
    long long total = (long long)E * (OUT_C / 4);
    k_edge_agg<OUT_C><<<(unsigned)((total + TPB - 1) / TPB), TPB, 0, stream>>>(
        h, norm, src, dst, out, E);
  }
  {
    long long total = (long long)N * (OUT_C / 4);
    k_node_finish<OUT_C, false><<<(unsigned)((total + TPB - 1) / TPB), TPB, 0, stream>>>(
        out, h, dinv, b2, N);
  }
}